// DSS_Base_34488587387072
// MI455X (gfx1250) — compile-verified
//
#include <hip/hip_runtime.h>

#define DIM 64

// Native clang vector type (accepted by __builtin_nontemporal_load/store).
typedef float vf2 __attribute__((ext_vector_type(2)));

// ---------------------------------------------------------------------------
// Zero a float region (workspace accumulators must start at 0 for atomics).
// ---------------------------------------------------------------------------
__global__ __launch_bounds__(256) void zero_ws(float4* __restrict__ p, long n4) {
    long i = (long)blockIdx.x * blockDim.x + threadIdx.x;
    if (i < n4) p[i] = make_float4(0.f, 0.f, 0.f, 0.f);
}

// ---------------------------------------------------------------------------
// COO SpMM scatter:  out[rows[e]] += vals[e] * src[cols[e]]
// One wave32 per edge (grid-strided). Lane l owns elements 2l, 2l+1 of the
// 64-float row: float2 gather (global_load_b64) + 2x global_atomic_add_f32.
// Edge arrays are streamed exactly once -> non-temporal loads (TH=NT) so they
// don't evict the L2-resident feature/destination tables. Source rows are
// reused across edges -> default RT caching + prefetch of the next edge's row
// (gfx1250 global_prefetch_b8).
// ---------------------------------------------------------------------------
__global__ __launch_bounds__(256)
void spmm_scatter(const int*   __restrict__ rows,
                  const int*   __restrict__ cols,
                  const float* __restrict__ vals,
                  int nnz,
                  const float* __restrict__ srcA,
                  const float* __restrict__ srcB,
                  int nA,
                  float* __restrict__ out) {
    const int lane   = threadIdx.x & 31;
    const int wave   = blockIdx.x * (blockDim.x >> 5) + (threadIdx.x >> 5);
    const int nwaves = gridDim.x * (blockDim.x >> 5);

    for (int e = wave; e < nnz; e += nwaves) {
        const int   r = __builtin_nontemporal_load(rows + e);
        const int   c = __builtin_nontemporal_load(cols + e);
        const float v = __builtin_nontemporal_load(vals + e);

        // Prefetch next iteration's randomly-indexed source row.
        const int en = e + nwaves;
        if (en < nnz) {
            const int cn = __builtin_nontemporal_load(cols + en);
            const float* pn = (cn < nA) ? (srcA + (long)cn * DIM)
                                        : (srcB + (long)(cn - nA) * DIM);
            __builtin_prefetch(pn + 2 * lane, 0, 1);
        }

        const float* src = (c < nA) ? (srcA + (long)c * DIM)
                                    : (srcB + (long)(c - nA) * DIM);
        const vf2 s = *(const vf2*)(src + 2 * lane);

        float* dst = out + (long)r * DIM + 2 * lane;
        atomicAdd(dst,     v * s.x);
        atomicAdd(dst + 1, v * s.y);
    }
}

// ---------------------------------------------------------------------------
// Per-row finalize: out = (feat + f1/max(||f1||,eps) + f2/max(||f2||,eps)) / 3
// One wave32 per node row; wave-wide shuffle reduction for the two norms.
// f1/f2 reads are last-use -> NT loads; output is write-once -> NT stores.
// ---------------------------------------------------------------------------
__global__ __launch_bounds__(256)
void finalize(const float* __restrict__ featA,
              const float* __restrict__ featB,
              int nA, int nTot,
              const float* __restrict__ f1,
              const float* __restrict__ f2,
              float* __restrict__ outA,
              float* __restrict__ outB) {
    const int lane = threadIdx.x & 31;
    const int row  = blockIdx.x * (blockDim.x >> 5) + (threadIdx.x >> 5);
    if (row >= nTot) return;

    const vf2 a = __builtin_nontemporal_load(
        (const vf2*)(f1 + (long)row * DIM + 2 * lane));
    const vf2 b = __builtin_nontemporal_load(
        (const vf2*)(f2 + (long)row * DIM + 2 * lane));

    float ss1 = a.x * a.x + a.y * a.y;
    float ss2 = b.x * b.x + b.y * b.y;
#pragma unroll
    for (int off = 16; off > 0; off >>= 1) {
        ss1 += __shfl_xor(ss1, off, 32);
        ss2 += __shfl_xor(ss2, off, 32);
    }
    const float inv1 = 1.0f / fmaxf(sqrtf(ss1), 1e-12f);
    const float inv2 = 1.0f / fmaxf(sqrtf(ss2), 1e-12f);

    const float* fp = (row < nA) ? (featA + (long)row * DIM)
                                 : (featB + (long)(row - nA) * DIM);
    const vf2 f = *(const vf2*)(fp + 2 * lane);

    float* op = (row < nA) ? (outA + (long)row * DIM)
                           : (outB + (long)(row - nA) * DIM);
    const float third = 1.0f / 3.0f;
    vf2 o;
    o.x = (f.x + a.x * inv1 + b.x * inv2) * third;
    o.y = (f.y + a.y * inv1 + b.y * inv2) * third;
    __builtin_nontemporal_store(o, (vf2*)(op + 2 * lane));
}

// ---------------------------------------------------------------------------
// Host-side orchestration.
// ---------------------------------------------------------------------------
extern "C" void kernel_launch(void* const* d_in, const int* in_sizes, int n_in,
                              void* d_out, int out_size, void* d_ws, size_t ws_size,
                              hipStream_t stream) {
    const float* users   = (const float*)d_in[0];
    const float* items   = (const float*)d_in[1];
    const float* bundles = (const float*)d_in[2];
    const float* ui_vals = (const float*)d_in[3];
    const float* bi_vals = (const float*)d_in[4];
    const float* ub_vals = (const float*)d_in[5];
    const int*   ui_rows = (const int*)d_in[6];
    const int*   ui_cols = (const int*)d_in[7];
    const int*   bi_rows = (const int*)d_in[8];
    const int*   bi_cols = (const int*)d_in[9];
    const int*   ub_rows = (const int*)d_in[10];
    const int*   ub_cols = (const int*)d_in[11];

    const int U  = in_sizes[0] / DIM;   // 100000
    const int NI = in_sizes[1] / DIM;   // 50000
    const int NB = in_sizes[2] / DIM;   // 20000
    const int nnz_ui = in_sizes[3];
    const int nnz_bi = in_sizes[4];
    const int nnz_ub = in_sizes[5];

    float* out = (float*)d_out;

    // Output slice bases (rows): [UI_u(U), UB_u(U), BI_b(NB), UB_b(NB), UI_i(NI), BI_i(NI)]
    float* UI_u = out;
    float* UB_u = out + (size_t)U * DIM;
    float* BI_b = out + (size_t)(2 * U) * DIM;
    float* UB_b = out + (size_t)(2 * U + NB) * DIM;
    float* UI_i = out + (size_t)(2 * U + 2 * NB) * DIM;
    float* BI_i = out + (size_t)(2 * U + 2 * NB + NI) * DIM;

    // Workspace: two contiguous feature buffers sized for the largest graph.
    int maxTot = U + NI;
    if (NB + NI > maxTot) maxTot = NB + NI;
    if (U + NB > maxTot)  maxTot = U + NB;
    float* f1 = (float*)d_ws;
    float* f2 = f1 + (size_t)maxTot * DIM;

    auto run = [&](const int* rows, const int* cols, const float* vals, int nnz,
                   const float* A, const float* B, int nA, int nB_,
                   float* outA, float* outB) {
        const int nTot = nA + nB_;

        // Zero f1 and f2 in a single launch: they are contiguous in the
        // workspace; covering the (small) gap when nTot < maxTot is harmless.
        {
            const long n4 = ((long)maxTot + nTot) * DIM / 4;
            int blk = 256;
            long grid = (n4 + blk - 1) / blk;
            zero_ws<<<(unsigned)grid, blk, 0, stream>>>((float4*)f1, n4);
        }

        const int wavesPerBlk  = 256 / 32;   // 8 waves per workgroup
        const int edgesPerWave = 4;

        // Layer 1: f1 = SpMM(edges, concat(A,B))
        {
            int blocks = (nnz + wavesPerBlk * edgesPerWave - 1) /
                         (wavesPerBlk * edgesPerWave);
            spmm_scatter<<<blocks, 256, 0, stream>>>(rows, cols, vals, nnz,
                                                     A, B, nA, f1);
        }

        // Layer 2: f2 = SpMM(edges, f1)   (source is the single contiguous f1)
        {
            int blocks = (nnz + wavesPerBlk * edgesPerWave - 1) /
                         (wavesPerBlk * edgesPerWave);
            spmm_scatter<<<blocks, 256, 0, stream>>>(rows, cols, vals, nnz,
                                                     f1, f1, nTot, f2);
        }

        // Finalize: (feat + l2norm(f1) + l2norm(f2)) / 3 -> output slices.
        {
            int blocks = (nTot + wavesPerBlk - 1) / wavesPerBlk;
            finalize<<<blocks, 256, 0, stream>>>(A, B, nA, nTot, f1, f2,
                                                 outA, outB);
        }
    };

    // UI graph: nodes = users + items
    run(ui_rows, ui_cols, ui_vals, nnz_ui, users, items, U, NI, UI_u, UI_i);
    // BI graph: nodes = bundles + items
    run(bi_rows, bi_cols, bi_vals, nnz_bi, bundles, items, NB, NI, BI_b, BI_i);
    // UB graph: nodes = users + bundles
    run(ub_rows, ub_cols, ub_vals, nnz_ub, users, bundles, U, NB, UB_u, UB_b);
}